// TinyGAT_21251498181386
// MI455X (gfx1250) — compile-verified
//
#include <hip/hip_runtime.h>

#define NN   100000
#define EE   3200000
#define ETOT (EE + NN)
#define NGG  512

typedef __attribute__((ext_vector_type(2))) float v2f;
typedef __attribute__((ext_vector_type(8))) float v8f;

// ---- monotone float <-> uint key for atomicMax-based segment max ----
__device__ __forceinline__ unsigned fkey(float f) {
    unsigned u = __float_as_uint(f);
    return (u & 0x80000000u) ? ~u : (u | 0x80000000u);
}
__device__ __forceinline__ float funkey(unsigned k) {
    return (k & 0x80000000u) ? __uint_as_float(k ^ 0x80000000u)
                             : __uint_as_float(~k);
}

// =====================================================================
// h = X @ W  (N x 32 = N x 32 * 32 x 32) via V_WMMA_F32_16X16X4_F32,
// fused with attention logits al_s[n,h], al_d[n,h].
// One wave handles a 16-row tile; 8 waves / block.
// =====================================================================
template <int H>
__global__ __launch_bounds__(256) void transform_kernel(
    const float* __restrict__ X, const float* __restrict__ W,
    const float* __restrict__ asrc, const float* __restrict__ adst,
    float* __restrict__ h, float* __restrict__ als, float* __restrict__ ald) {
    __shared__ float lds[8][16][33];
    const int wave = threadIdx.x >> 5;
    const int lane = threadIdx.x & 31;
    const int m0 = (blockIdx.x * 8 + wave) * 16;

    if (m0 < NN) {
        const int mrow = m0 + (lane & 15);
        const int kh = (lane >> 4) * 2;   // 0 or 2 (K-half per ISA A layout)
        const int col = lane & 15;
        v8f c0 = {}; v8f c1 = {};
#pragma unroll
        for (int k = 0; k < 32; k += 4) {
            v2f a;  a.x  = X[mrow * 32 + k + kh];
                    a.y  = X[mrow * 32 + k + kh + 1];
            v2f bf0; bf0.x = W[(k + kh) * 32 + col];
                     bf0.y = W[(k + kh + 1) * 32 + col];
            v2f bf1; bf1.x = W[(k + kh) * 32 + 16 + col];
                     bf1.y = W[(k + kh + 1) * 32 + 16 + col];
            c0 = __builtin_amdgcn_wmma_f32_16x16x4_f32(false, a, false, bf0,
                                                       (short)0, c0, false, false);
            c1 = __builtin_amdgcn_wmma_f32_16x16x4_f32(false, a, false, bf1,
                                                       (short)0, c1, false, false);
        }
        const int rb = (lane >> 4) * 8;   // C layout: VGPR j -> row j (+8 hi lanes)
#pragma unroll
        for (int j = 0; j < 8; ++j) {
            const int r = rb + j;
            lds[wave][r][col]      = c0[j];
            lds[wave][r][16 + col] = c1[j];
            h[(m0 + r) * 32 + col]      = c0[j];
            h[(m0 + r) * 32 + 16 + col] = c1[j];
        }
    }
    __syncthreads();
    if (m0 < NN) {
        const int r = lane >> 1;
        if (H == 4) {
            const int h0 = (lane & 1) * 2;
#pragma unroll
            for (int hh = h0; hh < h0 + 2; ++hh) {
                float ss = 0.f, sd = 0.f;
#pragma unroll
                for (int c = 0; c < 8; ++c) {
                    const float v = lds[wave][r][hh * 8 + c];
                    ss += v * asrc[hh * 8 + c];
                    sd += v * adst[hh * 8 + c];
                }
                als[(m0 + r) * 4 + hh] = ss;
                ald[(m0 + r) * 4 + hh] = sd;
            }
        } else if ((lane & 1) == 0) {
            float ss = 0.f, sd = 0.f;
#pragma unroll
            for (int c = 0; c < 32; ++c) {
                const float v = lds[wave][r][c];
                ss += v * asrc[c];
                sd += v * adst[c];
            }
            als[(m0 + r) * 4] = ss;
            ald[(m0 + r) * 4] = sd;
        }
    }
}

// ---- zero the per-layer accumulators (and pooling buffers) ----
__global__ void init_kernel(float* __restrict__ outb, unsigned* __restrict__ mU,
                            float* __restrict__ sb, float* __restrict__ pooled,
                            float* __restrict__ cnt) {
    const int i = blockIdx.x * blockDim.x + threadIdx.x;
    if (i < NN * 32) outb[i] = 0.f;
    if (i < NN * 4) { mU[i] = 0u; sb[i] = 0.f; }
    if (i < NGG * 32) pooled[i] = 0.f;
    if (i < NGG) cnt[i] = 0.f;
}

__device__ __forceinline__ void edge_sd(const int* __restrict__ ei, int t,
                                        int& s, int& d) {
    if (t < EE) { s = ei[t]; d = ei[EE + t]; }
    else        { s = t - EE; d = s; }           // self-loops appended
}

// ---- pass 1: segment max of leaky_relu(al_s[src]+al_d[dst]) over dst ----
template <int H>
__global__ void edge_max_kernel(const int* __restrict__ ei,
                                const float* __restrict__ als,
                                const float* __restrict__ ald,
                                unsigned* __restrict__ mU) {
    const int t = blockIdx.x * blockDim.x + threadIdx.x;
    if (t >= ETOT) return;
    __builtin_prefetch(ei + t + 8192, 0, 0);          // global_prefetch_b8
    int s, d; edge_sd(ei, t, s, d);
#pragma unroll
    for (int hh = 0; hh < H; ++hh) {
        float e = als[s * 4 + hh] + ald[d * 4 + hh];
        e = e > 0.f ? e : 0.2f * e;
        atomicMax(&mU[d * 4 + hh], fkey(e));
    }
}

// ---- pass 2: segment sum of exp(e - m[dst]) ----
template <int H>
__global__ void edge_expsum_kernel(const int* __restrict__ ei,
                                   const float* __restrict__ als,
                                   const float* __restrict__ ald,
                                   const unsigned* __restrict__ mU,
                                   float* __restrict__ sb) {
    const int t = blockIdx.x * blockDim.x + threadIdx.x;
    if (t >= ETOT) return;
    __builtin_prefetch(ei + t + 8192, 0, 0);
    int s, d; edge_sd(ei, t, s, d);
#pragma unroll
    for (int hh = 0; hh < H; ++hh) {
        float e = als[s * 4 + hh] + ald[d * 4 + hh];
        e = e > 0.f ? e : 0.2f * e;
        atomicAdd(&sb[d * 4 + hh], expf(e - funkey(mU[d * 4 + hh])));
    }
}

// ---- pass 3: out[dst] += alpha * h[src] ----
template <int H>
__global__ void edge_accum_kernel(const int* __restrict__ ei,
                                  const float* __restrict__ als,
                                  const float* __restrict__ ald,
                                  const unsigned* __restrict__ mU,
                                  const float* __restrict__ sb,
                                  const float* __restrict__ h,
                                  float* __restrict__ outb) {
    const int t = blockIdx.x * blockDim.x + threadIdx.x;
    if (t >= ETOT) return;
    __builtin_prefetch(ei + t + 8192, 0, 0);
    int s, d; edge_sd(ei, t, s, d);
    float wt[H];
#pragma unroll
    for (int hh = 0; hh < H; ++hh) {
        float e = als[s * 4 + hh] + ald[d * 4 + hh];
        e = e > 0.f ? e : 0.2f * e;
        wt[hh] = expf(e - funkey(mU[d * 4 + hh])) / sb[d * 4 + hh];
    }
    const float4* hs = (const float4*)(h + (long)s * 32);
    constexpr int SH = (H == 4) ? 3 : 5;   // channel -> head shift
#pragma unroll
    for (int q = 0; q < 8; ++q) {
        const float4 v = hs[q];
        const int base = q * 4;
        atomicAdd(&outb[d * 32 + base + 0], wt[(base + 0) >> SH] * v.x);
        atomicAdd(&outb[d * 32 + base + 1], wt[(base + 1) >> SH] * v.y);
        atomicAdd(&outb[d * 32 + base + 2], wt[(base + 2) >> SH] * v.z);
        atomicAdd(&outb[d * 32 + base + 3], wt[(base + 3) >> SH] * v.w);
    }
}

// ---- bias + optional ELU, in place ----
__global__ void finalize_kernel(float* __restrict__ outb,
                                const float* __restrict__ bias, int act) {
    const int i = blockIdx.x * blockDim.x + threadIdx.x;
    if (i >= NN * 32) return;
    float v = outb[i] + bias[i & 31];
    if (act) v = v > 0.f ? v : (expf(v) - 1.f);
    outb[i] = v;
}

// ---- global mean pool: accumulate sums + counts per graph ----
__global__ void pool_kernel(const float* __restrict__ outb,
                            const int* __restrict__ batch,
                            float* __restrict__ pooled, float* __restrict__ cnt) {
    const int n = blockIdx.x * blockDim.x + threadIdx.x;
    if (n >= NN) return;
    const int g = batch[n];
    const float4* p = (const float4*)(outb + (long)n * 32);
#pragma unroll
    for (int q = 0; q < 8; ++q) {
        const float4 v = p[q];
        atomicAdd(&pooled[g * 32 + q * 4 + 0], v.x);
        atomicAdd(&pooled[g * 32 + q * 4 + 1], v.y);
        atomicAdd(&pooled[g * 32 + q * 4 + 2], v.z);
        atomicAdd(&pooled[g * 32 + q * 4 + 3], v.w);
    }
    atomicAdd(&cnt[g], 1.0f);
}

// ---- [NG,32] @ [32,2] + b ----
__global__ void cls_kernel(const float* __restrict__ pooled,
                           const float* __restrict__ cnt,
                           const float* __restrict__ Wc,
                           const float* __restrict__ bc,
                           float* __restrict__ out) {
    const int g = blockIdx.x * blockDim.x + threadIdx.x;
    if (g >= NGG) return;
    const float inv = 1.f / fmaxf(cnt[g], 1.f);
    float a0 = bc[0], a1 = bc[1];
#pragma unroll
    for (int c = 0; c < 32; ++c) {
        const float p = pooled[g * 32 + c] * inv;
        a0 += p * Wc[c * 2 + 0];
        a1 += p * Wc[c * 2 + 1];
    }
    out[g * 2 + 0] = a0;
    out[g * 2 + 1] = a1;
}

extern "C" void kernel_launch(void* const* d_in, const int* in_sizes, int n_in,
                              void* d_out, int out_size, void* d_ws, size_t ws_size,
                              hipStream_t stream) {
    (void)in_sizes; (void)n_in; (void)out_size; (void)ws_size;
    const float* x     = (const float*)d_in[0];
    const int*   ei    = (const int*)d_in[1];
    const int*   batch = (const int*)d_in[2];
    const float* W0 = (const float*)d_in[3],  *as0 = (const float*)d_in[4];
    const float* ad0 = (const float*)d_in[5], *b0 = (const float*)d_in[6];
    const float* W1 = (const float*)d_in[7],  *as1 = (const float*)d_in[8];
    const float* ad1 = (const float*)d_in[9], *b1 = (const float*)d_in[10];
    const float* W2 = (const float*)d_in[11], *as2 = (const float*)d_in[12];
    const float* ad2 = (const float*)d_in[13],*b2 = (const float*)d_in[14];
    const float* clsW = (const float*)d_in[15],*clsb = (const float*)d_in[16];
    float* out = (float*)d_out;

    float*    ws     = (float*)d_ws;
    float*    hbuf   = ws;                      // N*32
    float*    outb   = hbuf + (long)NN * 32;    // N*32 (also next-layer input)
    float*    als    = outb + (long)NN * 32;    // N*4
    float*    ald    = als + NN * 4;            // N*4
    unsigned* mU     = (unsigned*)(ald + NN * 4); // N*4
    float*    sb     = (float*)(mU + NN * 4);   // N*4
    float*    pooled = sb + NN * 4;             // NG*32
    float*    cnt    = pooled + NGG * 32;       // NG

    const int TB = 256;
    const int gT = (NN / 16 + 7) / 8;
    const int gI = (NN * 32 + TB - 1) / TB;
    const int gE = (ETOT + TB - 1) / TB;
    const int gN = (NN + TB - 1) / TB;

    // ---------------- layer 0 (H=4, concat, ELU) ----------------
    transform_kernel<4><<<gT, TB, 0, stream>>>(x, W0, as0, ad0, hbuf, als, ald);
    init_kernel<<<gI, TB, 0, stream>>>(outb, mU, sb, pooled, cnt);
    edge_max_kernel<4><<<gE, TB, 0, stream>>>(ei, als, ald, mU);
    edge_expsum_kernel<4><<<gE, TB, 0, stream>>>(ei, als, ald, mU, sb);
    edge_accum_kernel<4><<<gE, TB, 0, stream>>>(ei, als, ald, mU, sb, hbuf, outb);
    finalize_kernel<<<gI, TB, 0, stream>>>(outb, b0, 1);

    // ---------------- layer 1 (H=4, concat, ELU) ----------------
    transform_kernel<4><<<gT, TB, 0, stream>>>(outb, W1, as1, ad1, hbuf, als, ald);
    init_kernel<<<gI, TB, 0, stream>>>(outb, mU, sb, pooled, cnt);
    edge_max_kernel<4><<<gE, TB, 0, stream>>>(ei, als, ald, mU);
    edge_expsum_kernel<4><<<gE, TB, 0, stream>>>(ei, als, ald, mU, sb);
    edge_accum_kernel<4><<<gE, TB, 0, stream>>>(ei, als, ald, mU, sb, hbuf, outb);
    finalize_kernel<<<gI, TB, 0, stream>>>(outb, b1, 1);

    // ---------------- layer 2 (H=1, mean over 1 head == identity) --------
    transform_kernel<1><<<gT, TB, 0, stream>>>(outb, W2, as2, ad2, hbuf, als, ald);
    init_kernel<<<gI, TB, 0, stream>>>(outb, mU, sb, pooled, cnt);
    edge_max_kernel<1><<<gE, TB, 0, stream>>>(ei, als, ald, mU);
    edge_expsum_kernel<1><<<gE, TB, 0, stream>>>(ei, als, ald, mU, sb);
    edge_accum_kernel<1><<<gE, TB, 0, stream>>>(ei, als, ald, mU, sb, hbuf, outb);
    finalize_kernel<<<gI, TB, 0, stream>>>(outb, b2, 0);

    // ---------------- pooling + classifier ----------------
    pool_kernel<<<gN, TB, 0, stream>>>(outb, batch, pooled, cnt);
    cls_kernel<<<(NGG + TB - 1) / TB, TB, 0, stream>>>(pooled, cnt, clsW, clsb, out);
}